// Kernel_1995877186436538400_71605694759413
// MI455X (gfx1250) — compile-verified
//
#include <hip/hip_runtime.h>

// ---- problem constants (from reference) ----
#define NN   16
#define CC   256
#define HH   56
#define WW   56
#define KK   7
#define DIL  3
#define PAD  9
#define HP   (HH + 2 * PAD)   // 74 padded rows
#define HW   (HH * WW)        // 3136
#define CHW  (CC * HW)        // 802816
#define IMG  (NN * CHW)       // 12845056
#define KDIM (CC * KK)        // 1792
#define T2CL_SZ (NN * HP * WW * CC)   // padded channels-last t2: 16973824 floats

typedef __attribute__((ext_vector_type(2))) float v2f;
typedef __attribute__((ext_vector_type(4))) float v4f;
typedef __attribute__((ext_vector_type(8))) float v8f;

// -------------------------------------------------------------------------
// Kernel 1: fused depthwise convs.
//   t1 (NCHW)                 = dwconv(x, w1[C,1,5,5], pad (2,2))
//   t2cl (padded channels-last) = dwconv(x, w2[C,1,3,1], pad (1,0))
//       t2cl[n][h+PAD][w][c]  (borders zero-filled by zero_pad_t2)
// -------------------------------------------------------------------------
__global__ void __launch_bounds__(256) prep_dw(const float* __restrict__ x,
                                               const float* __restrict__ w1,
                                               const float* __restrict__ w2,
                                               float* __restrict__ t1,
                                               float* __restrict__ t2cl) {
    int idx = blockIdx.x * 256 + threadIdx.x;
    if (idx >= IMG) return;
    int w = idx % WW;
    int t = idx / WW;
    int h = t % HH; t /= HH;
    int c = t % CC;
    int n = t / CC;
    const float* xb = x + (size_t)(n * CC + c) * HW;

    // t2: (3,1) kernel, pad (1,0) -> padded channels-last layout
    float s2 = 0.f;
#pragma unroll
    for (int r = 0; r < 3; ++r) {
        int hh = h + r - 1;
        if ((unsigned)hh < (unsigned)HH)
            s2 += w2[c * 3 + r] * xb[hh * WW + w];
    }
    t2cl[(size_t)((n * HP + h + PAD) * WW + w) * CC + c] = s2;

    // t1: 5x5 kernel, pad (2,2) -> NCHW (matches output indexing)
    float s1 = 0.f;
#pragma unroll
    for (int r = 0; r < 5; ++r) {
        int hh = h + r - 2;
        if ((unsigned)hh >= (unsigned)HH) continue;
#pragma unroll
        for (int s = 0; s < 5; ++s) {
            int ww2 = w + s - 2;
            if ((unsigned)ww2 < (unsigned)WW)
                s1 += w1[c * 25 + r * 5 + s] * xb[hh * WW + ww2];
        }
    }
    t1[idx] = s1;
}

// Zero the 9-row top/bottom borders of the padded channels-last t2.
__global__ void __launch_bounds__(256) zero_pad_t2(float* __restrict__ t2cl) {
    int idx = blockIdx.x * 256 + threadIdx.x;
    const int total = NN * 2 * PAD * WW * CC;
    if (idx >= total) return;
    int c = idx % CC;
    int t = idx / CC;
    int w = t % WW; t /= WW;
    int rr = t % (2 * PAD);
    int n = t / (2 * PAD);
    int hp = (rr < PAD) ? rr : rr + HH;   // rows [0,9) and [65,74)
    t2cl[(size_t)((n * HP + hp) * WW + w) * CC + c] = 0.f;
}

// -------------------------------------------------------------------------
// Kernel 2: re-layout w4 so K dimension is i' = k*256 + c (shift/mask decode;
// k is constant within each 256-wide K chunk of the GEMM).
//   w4t[o][k*256 + c] = w4[o][c*7 + k]
// -------------------------------------------------------------------------
__global__ void __launch_bounds__(256) transpose_w4(const float* __restrict__ w4,
                                                    float* __restrict__ w4t) {
    int idx = blockIdx.x * 256 + threadIdx.x;
    if (idx >= CC * KDIM) return;
    int i = idx % KDIM;        // destination K index: k*256 + c
    int o = idx / KDIM;
    int k = i >> 8;
    int c = i & 255;
    w4t[idx] = w4[o * KDIM + c * KK + k];
}

// -------------------------------------------------------------------------
// Kernel 3: GEMM (t4 = w4t x unfold(t2)) via V_WMMA_F32_16X16X4_F32, fused
// with the t1 gate. Workgroup = 8 waves; tile = 16(M) x 128(N); each wave
// owns one 16x16 D tile. A chunks (16 x 256) staged in LDS, shared by all
// waves. Padded channels-last t2 makes every B load an unconditional
// global_load_b64 with small immediate offsets (no exec masking, no branch).
//
// WMMA operand mapping (wave32):
//   A 16x4  f32: lane L, vgpr j -> A[M = L%16][Kf = 2*(L/16) + j]
//   B 4x16  f32: lane L, vgpr j -> B[Kf = 2*(L/16) + j][N = L%16]
//   C/D 16x16  : lane L, vgpr v -> D[M = v + 8*(L/16)][N = L%16]
// -------------------------------------------------------------------------
#define KCH 256                // K chunk (one unfold position k per chunk)
#define LDA 260                // padded LDS row stride (260 % 64 == 4 -> conflict-free)

__global__ void __launch_bounds__(256) gemm_gate(const float* __restrict__ w4t,
                                                 const float* __restrict__ t2cl,
                                                 const float* __restrict__ t1,
                                                 float* __restrict__ out) {
    __shared__ float a_lds[16 * LDA];   // 16.6 KB

    const int tid  = threadIdx.x;
    const int lane = tid & 31;
    const int wave = tid >> 5;
    const int half = lane >> 4;         // 0 or 1
    const int l16  = lane & 15;
    const int m_base = blockIdx.y * 16; // 16 output channels per block

    // Column (n,h,w) owned by this lane for the whole K loop.
    // 50176 total cols = 392 blocks * 8 waves * 16 lanes (exact).
    int q   = blockIdx.x * 128 + wave * 16 + l16;
    int n   = q / HW;                   // uniform per tile (HW % 16 == 0)
    int rem = q - n * HW;
    int h   = rem / WW;
    int w   = rem - h * WW;

    v8f acc = {};
    const float* ap0 = &a_lds[l16 * LDA + 2 * half];

    for (int kc0 = 0; kc0 < KDIM; kc0 += KCH) {
        // ---- cooperative A-chunk load: 16 rows x 256 cols = 1024 float4 ----
#pragma unroll
        for (int it = 0; it < 4; ++it) {
            int f   = tid + it * 256;   // float4 id, 0..1023
            int row = f >> 6;           // 64 float4 per row
            int c4  = f & 63;
            v4f av = *(const v4f*)(w4t + (size_t)(m_base + row) * KDIM + kc0 + c4 * 4);
            *(v4f*)(&a_lds[row * LDA + c4 * 4]) = av;
        }
        __syncthreads();

        // k constant over this chunk -> padded row hp = h + 3k in [0,73], always valid.
        const int k  = kc0 >> 8;
        const int hp = h + DIL * k;
        // B base: channels-last, this lane's (n,hp,w), channel offset 2*half.
        const float* bp = t2cl + (size_t)((n * HP + hp) * WW + w) * CC + 2 * half;

        // ---- 64 WMMA steps, software-pipelined one step ahead ----
        v2f a = *(const v2f*)(ap0);
        v2f b = *(const v2f*)(bp);
#pragma unroll 8
        for (int step = 0; step < KCH / 4 - 1; ++step) {
            v2f an = *(const v2f*)(ap0 + (step + 1) * 4);
            v2f bn = *(const v2f*)(bp + (step + 1) * 4);
            acc = __builtin_amdgcn_wmma_f32_16x16x4_f32(
                false, a, false, b, (short)0, acc, false, false);
            a = an;
            b = bn;
        }
        acc = __builtin_amdgcn_wmma_f32_16x16x4_f32(
            false, a, false, b, (short)0, acc, false, false);
        __syncthreads();
    }

    // ---- epilogue: gate with t1 and store (D layout above) ----
#pragma unroll
    for (int v = 0; v < 8; ++v) {
        int o = m_base + v + 8 * half;
        size_t oidx = (size_t)(n * CC + o) * HW + h * WW + w;
        out[oidx] = acc[v] * t1[oidx];
    }
}

// -------------------------------------------------------------------------
extern "C" void kernel_launch(void* const* d_in, const int* in_sizes, int n_in,
                              void* d_out, int out_size, void* d_ws, size_t ws_size,
                              hipStream_t stream) {
    const float* x  = (const float*)d_in[0];
    const float* w1 = (const float*)d_in[1];
    const float* w2 = (const float*)d_in[2];
    const float* w4 = (const float*)d_in[3];
    float* outp = (float*)d_out;

    float* ws   = (float*)d_ws;
    float* t1   = ws;                               // IMG floats
    float* t2cl = ws + (size_t)IMG;                 // T2CL_SZ floats (padded, channels-last)
    float* w4t  = t2cl + (size_t)T2CL_SZ;           // CC*KDIM floats

    zero_pad_t2<<<dim3((NN * 2 * PAD * WW * CC + 255) / 256), dim3(256), 0, stream>>>(t2cl);
    prep_dw<<<dim3((IMG + 255) / 256), dim3(256), 0, stream>>>(x, w1, w2, t1, t2cl);
    transpose_w4<<<dim3((CC * KDIM + 255) / 256), dim3(256), 0, stream>>>(w4, w4t);
    gemm_gate<<<dim3(392, 16), dim3(256), 0, stream>>>(w4t, t2cl, t1, outp);
}